// GatherUpdate_18597208392250
// MI455X (gfx1250) — compile-verified
//
#include <hip/hip_runtime.h>

// ---------------------------------------------------------------------------
// Problem constants (from reference): B=8, N_ATOMS=32768, N_RES=2048,
// C_S=384, C_ATOM=128.  M = B*N_RES = 16384 rows for the LN+GEMM.
// ---------------------------------------------------------------------------
#define CS      384
#define CATOM   128
#define NRES    2048
#define NATOMS  32768
#define NB      8
#define ROWS_PB 64          // rows of s per block
#define THREADS1 128        // 4 waves (wave32)

typedef __attribute__((ext_vector_type(2))) float v2f;
typedef __attribute__((ext_vector_type(4))) float v4f;
typedef __attribute__((ext_vector_type(8))) float v8f;

// ---------------------------------------------------------------------------
// Kernel 1: fused LayerNorm + fp32 WMMA GEMM.
//   upd[r, a] = sum_k LN(s)[r, k] * W[a, k]
// One block = 64 rows x 128 cols.  4 waves; wave w owns rows 16w..16w+15.
// A 16x4 f32 frag: lanes 0-15 = M rows, v0/v1 hold K=0/1 (K=2/3 for lanes>=16).
// B 4x16 f32 frag: lanes 0-15 = N cols, mirrored K striping.
// ---------------------------------------------------------------------------
__global__ __launch_bounds__(THREADS1)
void ln_wmma_gemm_kernel(const float* __restrict__ s,
                         const float* __restrict__ gamma,
                         const float* __restrict__ beta,
                         const float* __restrict__ W,     // [CATOM][CS] row-major
                         float* __restrict__ upd)         // [M][CATOM]
{
    __shared__ float lds_n[ROWS_PB * CS];   // 98304 B  (normed tile)
    __shared__ float lds_g[CS];
    __shared__ float lds_b[CS];
    __shared__ float lds_mu[ROWS_PB];
    __shared__ float lds_rs[ROWS_PB];

    const int  tid     = threadIdx.x;
    const long rowBase = (long)blockIdx.x * ROWS_PB;

    for (int i = tid; i < CS; i += THREADS1) {
        lds_g[i] = gamma[i];
        lds_b[i] = beta[i];
    }
    const float* sTile = s + rowBase * CS;
    for (int i = tid; i < ROWS_PB * CS; i += THREADS1)
        lds_n[i] = sTile[i];
    __syncthreads();

    // Row statistics: 2 lanes per row (pair within a wave), 192 elems each.
    {
        const int row  = tid >> 1;
        const int half = tid & 1;
        const float* r = &lds_n[row * CS + half * (CS / 2)];
        float sum = 0.f, sq = 0.f;
        #pragma unroll 4
        for (int j = 0; j < CS / 2; ++j) {
            float v = r[j];
            sum += v;
            sq  += v * v;
        }
        sum += __shfl_xor(sum, 1);
        sq  += __shfl_xor(sq, 1);
        if (half == 0) {
            float mu  = sum * (1.0f / CS);
            float var = sq * (1.0f / CS) - mu * mu;
            lds_mu[row] = mu;
            lds_rs[row] = rsqrtf(var + 1e-5f);
        }
    }
    __syncthreads();

    // Normalize in place.
    for (int i = tid; i < ROWS_PB * CS; i += THREADS1) {
        int row = i / CS;
        int col = i - row * CS;
        lds_n[i] = (lds_n[i] - lds_mu[row]) * lds_rs[row] * lds_g[col] + lds_b[col];
    }
    __syncthreads();

    // ---- WMMA GEMM: each wave -> 16 rows x 128 cols ----
    const int wave  = tid >> 5;
    const int lane  = tid & 31;
    const int mrow  = lane & 15;            // M (or N) index within frag
    const int khalf = (lane >> 4) << 1;     // 0 for lanes 0-15, 2 for 16-31
    const int r0    = wave * 16;

    v8f acc[8] = {};

    const float* nrow = &lds_n[(r0 + mrow) * CS];
    const float* wcol[8];
    #pragma unroll
    for (int n = 0; n < 8; ++n)
        wcol[n] = W + (long)(n * 16 + mrow) * CS + khalf;

    for (int k = 0; k < CS; k += 4) {
        v2f a;
        a.x = nrow[k + khalf];
        a.y = nrow[k + khalf + 1];
        #pragma unroll
        for (int n = 0; n < 8; ++n) {
            v2f b;
            b.x = wcol[n][k];
            b.y = wcol[n][k + 1];
            // 8 args: (neg_a, A, neg_b, B, c_mod, C, reuse_a, reuse_b)
            acc[n] = __builtin_amdgcn_wmma_f32_16x16x4_f32(
                false, a, false, b, (short)0, acc[n], false, false);
        }
    }

    // D layout: lanes 0-15 -> rows i (i=vgpr idx), lanes 16-31 -> rows i+8,
    // col = n*16 + (lane&15).
    const int   rOut    = (lane < 16) ? 0 : 8;
    float* updBase = upd + (rowBase + r0) * CATOM;
    #pragma unroll
    for (int n = 0; n < 8; ++n) {
        #pragma unroll
        for (int i = 0; i < 8; ++i) {
            updBase[(long)(i + rOut) * CATOM + n * 16 + mrow] = acc[n][i];
        }
    }
}

// ---------------------------------------------------------------------------
// Kernel 2: streaming gather + add (bandwidth bound, ~270 MB HBM).
// out[b,atom,:] = atom_embed[b,atom,:] + upd[b*NRES + idx[b,atom], :]
// float4 per thread; 32 threads per atom row share one index.
// atom_embed/out use non-temporal hints so upd (8.4 MB) stays hot in L2.
// ---------------------------------------------------------------------------
__global__ __launch_bounds__(256)
void gather_add_kernel(const v4f* __restrict__ atom_embed,
                       const v4f* __restrict__ upd,
                       const int* __restrict__ idx,
                       v4f* __restrict__ out,
                       long total4)
{
    long g = (long)blockIdx.x * blockDim.x + threadIdx.x;
    if (g >= total4) return;

    long atomRow = g >> 5;                 // / (CATOM/4)
    int  c4      = (int)(g & 31);
    int  b       = (int)(atomRow >> 15);   // / NATOMS
    int  r       = idx[atomRow];
    long updRow  = (long)b * NRES + r;

    v4f e = __builtin_nontemporal_load(&atom_embed[g]);
    v4f u = upd[updRow * 32 + c4];         // L2-resident gather
    __builtin_nontemporal_store(e + u, &out[g]);
}

// ---------------------------------------------------------------------------
// Inputs (setup_inputs order):
//   0: atom_embed [8,32768,128] f32   1: s [8,2048,384] f32
//   2: cond_to_s_idx [8,32768] int    3: ln_gamma [384] f32
//   4: ln_beta [384] f32              5: W [128,384] f32
// ---------------------------------------------------------------------------
extern "C" void kernel_launch(void* const* d_in, const int* in_sizes, int n_in,
                              void* d_out, int out_size, void* d_ws, size_t ws_size,
                              hipStream_t stream) {
    const float* atom_embed = (const float*)d_in[0];
    const float* s          = (const float*)d_in[1];
    const int*   idx        = (const int*)d_in[2];
    const float* gamma      = (const float*)d_in[3];
    const float* beta       = (const float*)d_in[4];
    const float* W          = (const float*)d_in[5];
    float*       out        = (float*)d_out;
    float*       upd        = (float*)d_ws;   // 16384*128*4 = 8 MB scratch

    const int M = NB * NRES;                  // 16384 rows
    ln_wmma_gemm_kernel<<<M / ROWS_PB, THREADS1, 0, stream>>>(
        s, gamma, beta, W, upd);

    const long total4 = (long)NB * NATOMS * (CATOM / 4);  // 8,388,608
    const int  blk2   = 256;
    const long grid2  = (total4 + blk2 - 1) / blk2;
    gather_add_kernel<<<(int)grid2, blk2, 0, stream>>>(
        (const v4f*)atom_embed, (const v4f*)upd, idx, (v4f*)out, total4);
}